// ChamferkNNDist_24790551233437
// MI455X (gfx1250) — compile-verified
//
#include <hip/hip_runtime.h>
#include <math.h>

// ---------------------------------------------------------------------------
// ChamferkNNDist for MI455X (gfx1250, wave32).
// Pairwise sq-dists via V_WMMA_F32_16X16X4_F32 with the norm folded into K=4:
//   A[m,:] = ( qx, qy, qz, |q|^2 )      (column points, M dim)
//   B[:,n] = (-2px,-2py,-2pz, 1 )       (row points,    N dim)
//   C[m][n] = d(q_m, p_n) - |p_n|^2     (row-constant offset added at end)
// C layout (16x16 f32, wave32): lane holds N = lane&15; VGPR r holds
// M = r + (lane>=16 ? 8 : 0)  -> per-lane private reductions over M,
// single shfl_xor(16) merge at the end.
//
// Column operands are staged ONCE per block into 64 KB of LDS as packed
// float4 (x,y,z,|q|^2); the hot loop is ds_load_b64 + v_wmma + min/top-6.
// ---------------------------------------------------------------------------

typedef __attribute__((ext_vector_type(2))) float v2f;
typedef __attribute__((ext_vector_type(8))) float v8f;

#define BATCH 8
#define NPTS 4096
#define TILE 16
#define NTILES (NPTS / TILE)                      // 256 column tiles
#define WAVES_PER_BLOCK 8
#define BLOCK_THREADS (WAVES_PER_BLOCK * 32)      // 256 threads = 8 waves
#define ROWTILE_BLOCKS (NTILES / WAVES_PER_BLOCK) // 32 blocks per batch
#define STAGE_CHUNKS (NPTS / BLOCK_THREADS)       // 16
#define ALPHA 1.05f
#define W_CHAMFER 5.0f
#define W_KNN 3.0f
#define FBIG 3.0e38f

// Sorted-insert of y into ascending t0..t5, dropping the largest.
#define INS6(xv)                                                     \
  do {                                                               \
    float y_ = (xv), m_;                                             \
    m_ = fminf(t0, y_); y_ = fmaxf(t0, y_); t0 = m_;                 \
    m_ = fminf(t1, y_); y_ = fmaxf(t1, y_); t1 = m_;                 \
    m_ = fminf(t2, y_); y_ = fmaxf(t2, y_); t2 = m_;                 \
    m_ = fminf(t3, y_); y_ = fmaxf(t3, y_); t3 = m_;                 \
    m_ = fminf(t4, y_); y_ = fmaxf(t4, y_); t4 = m_;                 \
    t5 = fminf(t5, y_);                                              \
  } while (0)

// Stage the whole column set for batch b into LDS as WMMA A-operands.
// cols[j] = (x, y, z, |q|^2).  Uniform control flow; all lanes active.
__device__ __forceinline__ void stage_cols(const float* __restrict__ src,
                                           float4* cols) {
#pragma unroll
  for (int k = 0; k < STAGE_CHUNKS; ++k) {
    const int idx = threadIdx.x + BLOCK_THREADS * k;
    if (k + 1 < STAGE_CHUNKS)                     // global_prefetch_b8
      __builtin_prefetch(&src[(size_t)(idx + BLOCK_THREADS) * 3], 0, 1);
    const float x = src[idx * 3 + 0];
    const float y = src[idx * 3 + 1];
    const float z = src[idx * 3 + 2];
    cols[idx] = make_float4(x, y, z, x * x + y * y + z * z);
  }
  __syncthreads();
}

// ---------------------------------------------------------------------------
// Kernel A: chamfer row minima.  rows = adv points, cols = ori points.
// chRow[b][i] = min_j ||adv_i - ori_j||^2
// ---------------------------------------------------------------------------
__global__ __launch_bounds__(BLOCK_THREADS)
void chamfer_rowmin_kernel(const float* __restrict__ adv,
                           const float* __restrict__ ori,
                           float* __restrict__ chRow) {
  __shared__ float4 cols[NPTS];                   // 64 KB LDS
  const int b    = blockIdx.x / ROWTILE_BLOCKS;
  const int rtb  = blockIdx.x % ROWTILE_BLOCKS;
  const int wave = threadIdx.x >> 5;
  const int lane = threadIdx.x & 31;
  const int half = lane >> 4;   // 0: K0/K1 half, 1: K2/K3 half
  const int l16  = lane & 15;

  const int i = (rtb * WAVES_PER_BLOCK + wave) * TILE + l16; // adv row idx

  const float* advB = adv + (size_t)b * NPTS * 3;
  const float* oriB = ori + (size_t)b * NPTS * 3;

  stage_cols(oriB, cols);

  // Row point -> B operand (K=4 x N=16); lanes 0-15: (K0,K1), 16-31: (K2,K3)
  const float px = advB[i * 3 + 0];
  const float py = advB[i * 3 + 1];
  const float pz = advB[i * 3 + 2];
  const float pn2 = px * px + py * py + pz * pz;
  v2f bmat;
  bmat.x = half ? (-2.0f * pz) : (-2.0f * px);
  bmat.y = half ? 1.0f         : (-2.0f * py);

  const v8f czero = {0.f, 0.f, 0.f, 0.f, 0.f, 0.f, 0.f, 0.f};
  float rm = FBIG;

  // Hot loop: ds_load_b64 (lane-selected half of packed float4) + WMMA + min.
  const float2* c2 = reinterpret_cast<const float2*>(cols);
  const int lofs = l16 * 2 + half;                // loop-invariant lane offset
#pragma unroll 4
  for (int jt = 0; jt < NTILES; ++jt) {
    const float2 aq = c2[jt * (TILE * 2) + lofs];
    v2f amat;
    amat.x = aq.x;  // lanes 0-15: x ; lanes 16-31: z
    amat.y = aq.y;  // lanes 0-15: y ; lanes 16-31: |q|^2

    // C[m][n] = d(q_m, p_n) - |p_n|^2  (EXEC all-1s: uniform flow)
    v8f c = __builtin_amdgcn_wmma_f32_16x16x4_f32(
        false, amat, false, bmat, (short)0, czero, false, false);

    rm = fminf(rm, fminf(fminf(fminf(c[0], c[1]), fminf(c[2], c[3])),
                         fminf(fminf(c[4], c[5]), fminf(c[6], c[7]))));
  }

  // merge M=0..7 half with M=8..15 half
  rm = fminf(rm, __shfl_xor(rm, 16, 32));
  if (half == 0) chRow[(size_t)b * NPTS + i] = rm + pn2;
}

// ---------------------------------------------------------------------------
// Kernel B: per-row 6 smallest of adv<->adv distances; drop smallest (self=0),
// value[b][i] = mean of the remaining 5.
// ---------------------------------------------------------------------------
__global__ __launch_bounds__(BLOCK_THREADS)
void knn_value_kernel(const float* __restrict__ adv,
                      float* __restrict__ value) {
  __shared__ float4 cols[NPTS];                   // 64 KB LDS
  const int b    = blockIdx.x / ROWTILE_BLOCKS;
  const int rtb  = blockIdx.x % ROWTILE_BLOCKS;
  const int wave = threadIdx.x >> 5;
  const int lane = threadIdx.x & 31;
  const int half = lane >> 4;
  const int l16  = lane & 15;

  const int i = (rtb * WAVES_PER_BLOCK + wave) * TILE + l16;

  const float* advB = adv + (size_t)b * NPTS * 3;

  stage_cols(advB, cols);

  const float px = advB[i * 3 + 0];
  const float py = advB[i * 3 + 1];
  const float pz = advB[i * 3 + 2];
  const float pn2 = px * px + py * py + pz * pz;
  v2f bmat;
  bmat.x = half ? (-2.0f * pz) : (-2.0f * px);
  bmat.y = half ? 1.0f         : (-2.0f * py);

  const v8f czero = {0.f, 0.f, 0.f, 0.f, 0.f, 0.f, 0.f, 0.f};
  float t0 = FBIG, t1 = FBIG, t2 = FBIG, t3 = FBIG, t4 = FBIG, t5 = FBIG;

  const float2* c2 = reinterpret_cast<const float2*>(cols);
  const int lofs = l16 * 2 + half;
#pragma unroll 2
  for (int jt = 0; jt < NTILES; ++jt) {
    const float2 aq = c2[jt * (TILE * 2) + lofs];
    v2f amat;
    amat.x = aq.x;
    amat.y = aq.y;

    v8f c = __builtin_amdgcn_wmma_f32_16x16x4_f32(
        false, amat, false, bmat, (short)0, czero, false, false);

    INS6(c[0]); INS6(c[1]); INS6(c[2]); INS6(c[3]);
    INS6(c[4]); INS6(c[5]); INS6(c[6]); INS6(c[7]);
  }

  // Merge the two halves' top-6 lists (read partner first, then insert).
  const float o0 = __shfl_xor(t0, 16, 32);
  const float o1 = __shfl_xor(t1, 16, 32);
  const float o2 = __shfl_xor(t2, 16, 32);
  const float o3 = __shfl_xor(t3, 16, 32);
  const float o4 = __shfl_xor(t4, 16, 32);
  const float o5 = __shfl_xor(t5, 16, 32);
  INS6(o0); INS6(o1); INS6(o2); INS6(o3); INS6(o4); INS6(o5);

  // t0 is the self-distance (d=0 -> stored value -|p|^2): dropped, as in ref.
  if (half == 0)
    value[(size_t)b * NPTS + i] = 0.2f * (t1 + t2 + t3 + t4 + t5) + pn2;
}

// ---------------------------------------------------------------------------
// Kernel C: per-batch stats (deterministic fixed-order reductions).
// ---------------------------------------------------------------------------
__device__ __forceinline__ float block_sum_256(float v) {
  __shared__ float red[WAVES_PER_BLOCK];
  for (int off = 16; off > 0; off >>= 1) v += __shfl_xor(v, off, 32);
  __syncthreads();  // protect red across successive calls
  if ((threadIdx.x & 31) == 0) red[threadIdx.x >> 5] = v;
  __syncthreads();
  float s = red[0];
#pragma unroll
  for (int w = 1; w < WAVES_PER_BLOCK; ++w) s += red[w];
  return s;  // full sum, every thread
}

__global__ __launch_bounds__(256)
void batch_stats_kernel(const float* __restrict__ chRow,
                        const float* __restrict__ value,
                        float* __restrict__ perb) {
  const int b   = blockIdx.x;
  const int tid = threadIdx.x;
  const float* ch = chRow + (size_t)b * NPTS;
  const float* vv = value + (size_t)b * NPTS;

  float vloc[NPTS / 256];
  float sc = 0.f, sv = 0.f, sq = 0.f;
#pragma unroll
  for (int k = 0; k < NPTS / 256; ++k) {
    const int idx = tid + 256 * k;
    sc += ch[idx];
    const float v = vv[idx];
    vloc[k] = v;
    sv += v;
    sq += v * v;
  }
  const float Sc = block_sum_256(sc);
  const float Sv = block_sum_256(sv);
  const float Sq = block_sum_256(sq);

  const float mean = Sv / (float)NPTS;
  const float var  = (Sq - Sv * Sv / (float)NPTS) / (float)(NPTS - 1); // ddof=1
  const float thr  = mean + ALPHA * sqrtf(fmaxf(var, 0.f));

  float ms = 0.f;
#pragma unroll
  for (int k = 0; k < NPTS / 256; ++k)
    if (vloc[k] > thr) ms += vloc[k];
  const float Ms = block_sum_256(ms);

  if (tid == 0) {
    perb[2 * b + 0] = Sc / (float)NPTS;  // chamfer_per_b
    perb[2 * b + 1] = Ms / (float)NPTS;  // knn_per_b (masked mean)
  }
}

// ---------------------------------------------------------------------------
// Kernel D: scalar combine.
// ---------------------------------------------------------------------------
__global__ void finalize_kernel(const float* __restrict__ perb,
                                float* __restrict__ out) {
  if (threadIdx.x == 0 && blockIdx.x == 0) {
    float chs = 0.f, kns = 0.f;
#pragma unroll
    for (int b = 0; b < BATCH; ++b) {
      chs += perb[2 * b + 0];
      kns += perb[2 * b + 1];
    }
    out[0] = W_CHAMFER * (chs / (float)BATCH) + W_KNN * (kns / (float)BATCH);
  }
}

// ---------------------------------------------------------------------------
extern "C" void kernel_launch(void* const* d_in, const int* in_sizes, int n_in,
                              void* d_out, int out_size, void* d_ws,
                              size_t ws_size, hipStream_t stream) {
  const float* adv = (const float*)d_in[0];  // [8,4096,3] f32
  const float* ori = (const float*)d_in[1];  // [8,4096,3] f32
  float* ws = (float*)d_ws;
  float* chRow = ws;                               // BATCH*NPTS floats
  float* value = ws + (size_t)BATCH * NPTS;        // BATCH*NPTS floats
  float* perb  = ws + (size_t)2 * BATCH * NPTS;    // 2*BATCH floats

  const dim3 grid(BATCH * ROWTILE_BLOCKS);
  chamfer_rowmin_kernel<<<grid, BLOCK_THREADS, 0, stream>>>(adv, ori, chRow);
  knn_value_kernel<<<grid, BLOCK_THREADS, 0, stream>>>(adv, value);
  batch_stats_kernel<<<dim3(BATCH), 256, 0, stream>>>(chRow, value, perb);
  finalize_kernel<<<dim3(1), 32, 0, stream>>>(perb, (float*)d_out);
}